// Encoder_33930241638507
// MI455X (gfx1250) — compile-verified
//
#include <hip/hip_runtime.h>

#define N_NODES 10000
#define N_EDGES 100000
#define T_NODE  32
#define T_OP    16
#define H       64
#define HALF    32
#define GCOLS   544   // 512 basis columns (t*32+o) + 32 bias-projection columns

typedef __attribute__((ext_vector_type(2))) float v2f;
typedef __attribute__((ext_vector_type(8))) float v8f;

// ---------------------------------------------------------------------------
// One 16x16 output tile of C += A(MxK,row-major) @ B(KxN,row-major) using
// V_WMMA_F32_16X16X4_F32.  Per ISA 7.12.2:
//   A 16x4 f32 : lane<16 -> {K=0,K=1}, lane>=16 -> {K=2,K=3}, M = lane&15
//   B 4x16 f32 : mirrored, N = lane&15
//   C/D 16x16  : VGPR v -> M = v + 8*(lane>=16), N = lane&15
// ---------------------------------------------------------------------------
__device__ __forceinline__ v8f wmma_f32_tile(const float* __restrict__ A, int lda,
                                             const float* __restrict__ B, int ldb,
                                             int m0, int n0, int K, v8f acc, int lane) {
  const int mn = lane & 15;          // A row / B col for this lane
  const int kh = (lane >> 4) << 1;   // 0 for lanes 0-15, 2 for lanes 16-31
  for (int k0 = 0; k0 < K; k0 += 4) {
    v2f a, b;
    a.x = A[(m0 + mn) * lda + (k0 + kh + 0)];
    a.y = A[(m0 + mn) * lda + (k0 + kh + 1)];
    b.x = B[(k0 + kh + 0) * ldb + (n0 + mn)];
    b.y = B[(k0 + kh + 1) * ldb + (n0 + mn)];
    acc = __builtin_amdgcn_wmma_f32_16x16x4_f32(false, a, false, b,
                                                (short)0, acc, false, false);
  }
  return acc;
}

__device__ __forceinline__ void wmma_store(float* __restrict__ C, int ldc,
                                           int m0, int n0, v8f c, int lane) {
  const int n  = lane & 15;
  const int mb = (lane >> 4) << 3;
  #pragma unroll
  for (int v = 0; v < 8; ++v)
    C[(size_t)(m0 + mb + v) * ldc + (n0 + n)] = c[v];
}

// h0 = x (10000x32) @ node_init_w (32x64)
__global__ void k_init_gemm(const float* __restrict__ x, const float* __restrict__ w,
                            float* __restrict__ h) {
  int wid  = blockIdx.x * (blockDim.x >> 5) + (threadIdx.x >> 5);
  int lane = threadIdx.x & 31;
  if (wid >= 625 * 4) return;
  int mt = wid >> 2, nt = wid & 3;
  v8f acc = {};
  acc = wmma_f32_tile(x, T_NODE, w, H, mt * 16, nt * 16, T_NODE, acc, lane);
  wmma_store(h, H, mt * 16, nt * 16, acc, lane);
}

// hp = relu(layer_norm(h))  — one wave32 per node, shfl_xor reductions
__global__ void k_ln_relu(const float* __restrict__ h, const float* __restrict__ g,
                          const float* __restrict__ bta, float* __restrict__ hp) {
  int node = blockIdx.x * (blockDim.x >> 5) + (threadIdx.x >> 5);
  int lane = threadIdx.x & 31;
  if (node >= N_NODES) return;
  float a0 = h[(size_t)node * H + lane];
  float a1 = h[(size_t)node * H + lane + 32];
  float s = a0 + a1, q = a0 * a0 + a1 * a1;
  #pragma unroll
  for (int off = 16; off > 0; off >>= 1) {
    s += __shfl_xor(s, off, 32);
    q += __shfl_xor(q, off, 32);
  }
  float m   = s * (1.0f / H);
  float var = q * (1.0f / H) - m * m;
  float inv = rsqrtf(var + 1e-5f);
  float o0 = (a0 - m) * inv * g[lane] + bta[lane];
  float o1 = (a1 - m) * inv * g[lane + 32] + bta[lane + 32];
  hp[(size_t)node * H + lane]      = fmaxf(o0, 0.0f);
  hp[(size_t)node * H + lane + 32] = fmaxf(o1, 0.0f);
}

// G[i, c] : c<512 -> ew[t*2048 + i*32 + o] with t=c>>5, o=c&31 ; c>=512 -> eb[i*32 + (c-512)]
__global__ void k_build_G(const float* __restrict__ ew, const float* __restrict__ eb,
                          float* __restrict__ G) {
  int idx = blockIdx.x * blockDim.x + threadIdx.x;
  if (idx >= H * GCOLS) return;
  int i = idx / GCOLS, c = idx - i * GCOLS;
  float v;
  if (c < 512) {
    int t = c >> 5, o = c & 31;
    v = ew[t * (H * HALF) + i * HALF + o];
  } else {
    v = eb[i * HALF + (c - 512)];
  }
  G[idx] = v;
}

// M (10000x544) = hp (10000x64) @ G (64x544)   — the basis projection
__global__ void k_basis_gemm(const float* __restrict__ hp, const float* __restrict__ G,
                             float* __restrict__ M) {
  int wid  = blockIdx.x * (blockDim.x >> 5) + (threadIdx.x >> 5);
  int lane = threadIdx.x & 31;
  if (wid >= 625 * 34) return;
  int mt = wid / 34, nt = wid % 34;
  v8f acc = {};
  acc = wmma_f32_tile(hp, H, G, GCOLS, mt * 16, nt * 16, H, acc, lane);
  wmma_store(M, GCOLS, mt * 16, nt * 16, acc, lane);
}

// hnext[:, b*32 .. b*32+32) = h + hp @ root_w + root_b   (accumulator pre-seeded)
__global__ void k_root_gemm(const float* __restrict__ hp, const float* __restrict__ hcur,
                            const float* __restrict__ rw, const float* __restrict__ rb,
                            float* __restrict__ hnext, int branch) {
  int wid  = blockIdx.x * (blockDim.x >> 5) + (threadIdx.x >> 5);
  int lane = threadIdx.x & 31;
  if (wid >= 625 * 2) return;
  int mt = wid >> 1, nt = wid & 1;
  int m0 = mt * 16, n0 = nt * 16;
  int n = lane & 15, mb = (lane >> 4) << 3;
  v8f acc;
  #pragma unroll
  for (int v = 0; v < 8; ++v)
    acc[v] = hcur[(size_t)(m0 + mb + v) * H + branch * HALF + n0 + n] + rb[n0 + n];
  acc = wmma_f32_tile(hp, H, rw, HALF, m0, n0, H, acc, lane);
  #pragma unroll
  for (int v = 0; v < 8; ++v)
    hnext[(size_t)(m0 + mb + v) * H + branch * HALF + n0 + n] = acc[v];
}

// Per edge (one wave32, lane = output channel o):
//   msg[o] = M[src, 512+o] + sum_t ea[eai,t] * M[src, t*32+o]
//   atomic add into hnext[dst, b*32+o]
// Branch 1 re-pairs endpoints j with edge_attr[E-1-j] (reversed edge list).
__global__ void k_edge_msg(const float* __restrict__ M, const float* __restrict__ ea,
                           const long long* __restrict__ ei, float* __restrict__ hnext,
                           int branch) {
  int e    = blockIdx.x * (blockDim.x >> 5) + (threadIdx.x >> 5);
  int lane = threadIdx.x & 31;
  if (e >= N_EDGES) return;
  int eai = branch ? (N_EDGES - 1 - e) : e;
  int src = (int)ei[e];
  int dst = (int)ei[N_EDGES + e];
  const float* mrow = M + (size_t)src * GCOLS;
  float w[T_OP];
  #pragma unroll
  for (int t = 0; t < T_OP; ++t) w[t] = ea[(size_t)eai * T_OP + t];
  float msg = mrow[512 + lane];
  #pragma unroll
  for (int t = 0; t < T_OP; ++t) msg = fmaf(w[t], mrow[t * 32 + lane], msg);
  unsafeAtomicAdd(&hnext[(size_t)dst * H + branch * HALF + lane], msg);
}

extern "C" void kernel_launch(void* const* d_in, const int* in_sizes, int n_in,
                              void* d_out, int out_size, void* d_ws, size_t ws_size,
                              hipStream_t stream) {
  const float*     x      = (const float*)d_in[0];
  const long long* ei     = (const long long*)d_in[1];
  const float*     ea     = (const float*)d_in[2];
  const float*     w0     = (const float*)d_in[3];
  const float*     ln_g   = (const float*)d_in[4];
  const float*     ln_b   = (const float*)d_in[5];
  const float*     edge_w = (const float*)d_in[6];
  const float*     edge_b = (const float*)d_in[7];
  const float*     root_w = (const float*)d_in[8];
  const float*     root_b = (const float*)d_in[9];
  float* out = (float*)d_out;

  float* h0 = (float*)d_ws;                 // 10000*64
  float* h1 = h0 + (size_t)N_NODES * H;     // 10000*64
  float* hp = h1 + (size_t)N_NODES * H;     // 10000*64
  float* G  = hp + (size_t)N_NODES * H;     // 64*544
  float* M  = G  + (size_t)H * GCOLS;       // 10000*544  (~29.6 MB total)

  k_init_gemm<<<625, 128, 0, stream>>>(x, w0, h0);

  for (int l = 0; l < 3; ++l) {
    float* hc = (l == 1) ? h1 : h0;
    float* hn = (l == 0) ? h1 : ((l == 1) ? h0 : out);
    for (int b = 0; b < 2; ++b) {
      int lb = l * 2 + b;
      k_ln_relu<<<1250, 256, 0, stream>>>(hc, ln_g + (size_t)lb * H,
                                          ln_b + (size_t)lb * H, hp);
      k_build_G<<<(H * GCOLS + 255) / 256, 256, 0, stream>>>(
          edge_w + (size_t)lb * T_OP * H * HALF,
          edge_b + (size_t)lb * H * HALF, G);
      k_basis_gemm<<<(625 * 34 + 7) / 8, 256, 0, stream>>>(hp, G, M);
      k_root_gemm<<<(625 * 2 + 7) / 8, 256, 0, stream>>>(
          hp, hc, root_w + (size_t)lb * H * HALF,
          root_b + (size_t)lb * HALF, hn, b);
      k_edge_msg<<<N_EDGES / 8, 256, 0, stream>>>(M, ea, ei, hn, b);
    }
  }
}